// SpatialAttention_54065048322469
// MI455X (gfx1250) — compile-verified
//
#include <hip/hip_runtime.h>

typedef __attribute__((ext_vector_type(2))) float v2f;
typedef __attribute__((ext_vector_type(8))) float v8f;

#define NN 256
#define FF 192   // C*T
#define MROWS 2048  // B*N

__device__ __forceinline__ float sigmoid_f(float x) {
    // sigmoid(x) = 1/(1+e^-x) = rcp(1 + exp2(-x*log2(e)))  -- v_exp_f32/v_rcp_f32
    return __builtin_amdgcn_rcpf(1.0f + __builtin_amdgcn_exp2f(-1.44269504088896f * x));
}

// ---------------- Kernel 1: lhs/rhs GEMM via V_WMMA_F32_16X16X4_F32 ----------------
// D[i][n] = sum_f X[i][f] * W[n][f];  one wave per 16x16 tile, K swept 4 at a time.
__global__ __launch_bounds__(256) void gemm_wmma(const float* __restrict__ X,
                                                 const float* __restrict__ W1,
                                                 const float* __restrict__ W2,
                                                 float* __restrict__ ws) {
    const int wave = (blockIdx.x << 3) + (threadIdx.x >> 5);   // 0..4095 (wave-uniform)
    const int lane = threadIdx.x & 31;
    const int sel  = wave >> 11;            // 0 -> lhs (W1), 1 -> rhs (W2)
    const int rem  = wave & 2047;
    const int m0   = (rem >> 4) << 4;       // row tile in B*N  (0..2032)
    const int n0   = (rem & 15) << 4;       // col tile in N    (0..240)
    const float* W = sel ? W2 : W1;
    float* out = ws + (size_t)sel * ((size_t)MROWS * NN);

    // A 16x4 f32 layout: lanes 0-15 M=0..15 K={0,1}; lanes 16-31 K={2,3}
    const int mrow = m0 + (lane & 15);
    const int ncol = n0 + (lane & 15);
    const int koff = (lane >> 4) << 1;

    const float* Arow = X + (size_t)mrow * FF + koff;
    const float* Brow = W + (size_t)ncol * FF + koff;

    v8f acc = {};
#pragma unroll
    for (int f = 0; f < FF; f += 4) {
        v2f a = *(const v2f*)(Arow + f);
        v2f b = *(const v2f*)(Brow + f);
        acc = __builtin_amdgcn_wmma_f32_16x16x4_f32(
            /*neg_a=*/false, a, /*neg_b=*/false, b,
            /*c_mod=*/(short)0, acc, /*reuse_a=*/false, /*reuse_b=*/false);
    }

    // C/D 16x16 f32: VGPR v -> M = v + 8*(lane>=16), N = lane&15
    const int rbase = m0 + ((lane >> 4) << 3);
    const int col   = n0 + (lane & 15);
#pragma unroll
    for (int v = 0; v < 8; ++v) {
        out[(size_t)(rbase + v) * NN + col] = acc[v];
    }
}

// ---------------- Kernel 2: S[b,i,j] = sum_k sigmoid(lhs[b,i,k]+rhs[b,j,k]) + Vs ----
// 32x32 (i,j) tile per block; LDS holds both 32x256 slices transposed ([k][row]).
__global__ __launch_bounds__(256) void pair_sig(const float* __restrict__ ws,
                                                const float* __restrict__ Vs,
                                                float* __restrict__ S) {
    __shared__ float Lt[NN][32];   // [k][i]  32 KB
    __shared__ float Rt[NN][32];   // [k][j]  32 KB
    const float* lhs = ws;
    const float* rhs = ws + (size_t)MROWS * NN;
    const int b  = blockIdx.z;
    const int i0 = blockIdx.y << 5;
    const int j0 = blockIdx.x << 5;
    const int tid = threadIdx.x;
    const float* Lg = lhs + (size_t)(b * NN + i0) * NN;
    const float* Rg = rhs + (size_t)(b * NN + j0) * NN;

#pragma unroll
    for (int c = 0; c < 8; ++c) {
        int e  = tid + (c << 8);      // 0..2047 float4 slots over 32x256
        int r  = e >> 6;              // 0..31
        int c4 = (e & 63) << 2;       // 0..252
        float4 lv = *(const float4*)(Lg + r * NN + c4);
        float4 rv = *(const float4*)(Rg + r * NN + c4);
        Lt[c4+0][r] = lv.x; Lt[c4+1][r] = lv.y; Lt[c4+2][r] = lv.z; Lt[c4+3][r] = lv.w;
        Rt[c4+0][r] = rv.x; Rt[c4+1][r] = rv.y; Rt[c4+2][r] = rv.z; Rt[c4+3][r] = rv.w;
    }
    __syncthreads();

    const int tx = tid & 15, ty = tid >> 4;
    const int i = ty << 1, j = tx << 1;       // 2x2 micro-tile
    float s00 = 0.f, s01 = 0.f, s10 = 0.f, s11 = 0.f;
#pragma unroll 4
    for (int k = 0; k < NN; ++k) {
        float l0 = Lt[k][i], l1 = Lt[k][i + 1];
        float r0 = Rt[k][j], r1 = Rt[k][j + 1];
        s00 += sigmoid_f(l0 + r0);
        s01 += sigmoid_f(l0 + r1);
        s10 += sigmoid_f(l1 + r0);
        s11 += sigmoid_f(l1 + r1);
    }

    const int gi = i0 + i, gj = j0 + j;
    float*       Srow = S  + ((size_t)b * NN + gi) * NN + gj;
    const float* Vrow = Vs + (size_t)gi * NN + gj;
    Srow[0]      = s00 + Vrow[0];
    Srow[1]      = s01 + Vrow[1];
    Srow[NN]     = s10 + Vrow[NN];
    Srow[NN + 1] = s11 + Vrow[NN + 1];
}

// ---------------- Kernel 3: row softmax, one wave32 per row, in place ----------------
__global__ __launch_bounds__(256) void softmax_rows(float* __restrict__ S) {
    const int row  = (blockIdx.x << 3) + (threadIdx.x >> 5);  // 0..2047
    const int lane = threadIdx.x & 31;
    float* base = S + (size_t)row * NN;
    float v[8];
#pragma unroll
    for (int q = 0; q < 8; ++q) v[q] = base[lane + (q << 5)];
    float m = v[0];
#pragma unroll
    for (int q = 1; q < 8; ++q) m = fmaxf(m, v[q]);
#pragma unroll
    for (int off = 16; off > 0; off >>= 1) m = fmaxf(m, __shfl_xor(m, off, 32));
    float sum = 0.f;
#pragma unroll
    for (int q = 0; q < 8; ++q) {
        v[q] = __builtin_amdgcn_exp2f(1.44269504088896f * (v[q] - m));
        sum += v[q];
    }
#pragma unroll
    for (int off = 16; off > 0; off >>= 1) sum += __shfl_xor(sum, off, 32);
    const float inv = 1.0f / sum;
#pragma unroll
    for (int q = 0; q < 8; ++q) base[lane + (q << 5)] = v[q] * inv;
}

extern "C" void kernel_launch(void* const* d_in, const int* in_sizes, int n_in,
                              void* d_out, int out_size, void* d_ws, size_t ws_size,
                              hipStream_t stream) {
    (void)in_sizes; (void)n_in; (void)out_size; (void)ws_size;
    const float* X  = (const float*)d_in[0];   // (B,N,C,T) contiguous == (2048,192)
    const float* W1 = (const float*)d_in[1];   // (256,192)
    const float* W2 = (const float*)d_in[2];   // (256,192)
    const float* Vs = (const float*)d_in[3];   // (256,256)
    float* S  = (float*)d_out;                 // (8,256,256)
    float* ws = (float*)d_ws;                  // lhs @ 0, rhs @ 2MB (needs 4MB)

    gemm_wmma<<<512, 256, 0, stream>>>(X, W1, W2, ws);
    dim3 g2(8, 8, 8);                          // (j-tiles, i-tiles, B)
    pair_sig<<<g2, 256, 0, stream>>>(ws, Vs, S);
    softmax_rows<<<256, 256, 0, stream>>>(S);
}